// ScaledDotProductAttention_23905787969979
// MI455X (gfx1250) — compile-verified
//
#include <hip/hip_runtime.h>
#include <math.h>

// ---------------------------------------------------------------------------
// Fused attention for MI455X (gfx1250, wave32):
//   scores = Q K^T / 8  -> LDS (32 rows x 2048 cols, 256KB strip per WGP)
//   softmax rows, multiply structured group-weight matrix, write attn once
//   output = attn @ V via fp32 WMMA, attn consumed from LDS
// Matrix math: V_WMMA_F32_16X16X4_F32 (full fp32 precision, both outputs are
// fp32-compared).
// ---------------------------------------------------------------------------

typedef float v2f __attribute__((ext_vector_type(2)));
typedef float v8f __attribute__((ext_vector_type(8)));

#define B_DIM       32
#define S_DIM       2048
#define D_DIM       64
#define ROWS_PER_WG 32
#define LDS_STRIDE  2050   // floats/row; 2050 % 8 == 2 -> conflict-free tile
                           // stores (8*2050 % 64 == 16) and cheap phase-3 reads

extern "C" __global__ __launch_bounds__(256)
void attn_fused_kernel(const float* __restrict__ q,
                       const float* __restrict__ k,
                       const float* __restrict__ v,
                       const float* __restrict__ attn_w,
                       const int*   __restrict__ kcls_p,
                       float* __restrict__ outp,
                       float* __restrict__ attnp)
{
    extern __shared__ float smem[];                 // scores + row stats
    float* sc      = smem;                          // [32][LDS_STRIDE]
    float* rowmax  = smem + ROWS_PER_WG * LDS_STRIDE;
    float* rowrsum = rowmax + ROWS_PER_WG;

    const int tid  = threadIdx.x;
    const int lane = tid & 31;
    const int wave = tid >> 5;        // 0..7
    const int half = lane >> 4;       // 0/1  (lane groups per WMMA layout)
    const int l16  = lane & 15;

    const int bidx = blockIdx.y;      // batch
    const int r0   = blockIdx.x * ROWS_PER_WG;

    const int   Kcls = kcls_p[0];     // class size (K=32)
    const float w0 = tanhf(attn_w[0]);
    const float w1 = tanhf(attn_w[1]);
    const float w2 = tanhf(attn_w[2]);

    // ---------------- Phase 1: scores = (Q K^T) / 8 -> LDS ----------------
    {
        const int rt  = wave >> 2;                    // row tile 0..1
        const int row = r0 + rt * 16 + l16;
        const float* qrow = q + ((size_t)bidx * S_DIM + row) * D_DIM;

        // A operands (16x4 f32 layout): lane holds M=l16, K = 4*kk + 2*half + {0,1}
        v2f a[16];
        #pragma unroll
        for (int kk = 0; kk < 16; ++kk)
            a[kk] = *(const v2f*)(qrow + 4 * kk + 2 * half);

        for (int ct = (wave & 3); ct < S_DIM / 16; ct += 4) {
            const int col = ct * 16 + l16;
            const float* krow = k + ((size_t)bidx * S_DIM + col) * D_DIM;
            v8f c = {};
            #pragma unroll
            for (int kk = 0; kk < 16; ++kk) {
                // B (4x16 f32): lane holds N=l16, K-rows 2*half + {0,1}
                v2f b = *(const v2f*)(krow + 4 * kk + 2 * half);
                c = __builtin_amdgcn_wmma_f32_16x16x4_f32(
                        false, a[kk], false, b, (short)0, c, false, false);
            }
            // C/D layout: vgpr i, lane -> M = i + 8*half, N = l16
            #pragma unroll
            for (int i = 0; i < 8; ++i) {
                int m = rt * 16 + i + 8 * half;
                sc[m * LDS_STRIDE + ct * 16 + l16] = c[i] * 0.125f;  // /TEMPERATURE
            }
        }
    }
    __syncthreads();

    // ---------------- Phase 2a: per-row max and exp-sum ----------------
    {
        const int r   = tid >> 3;     // 0..31, 8 lanes per row (same wave)
        const int sub = tid & 7;
        const float* row = sc + r * LDS_STRIDE;
        float m = -3.402823466e38f;
        for (int c = sub; c < S_DIM; c += 8) m = fmaxf(m, row[c]);
        m = fmaxf(m, __shfl_xor(m, 1));
        m = fmaxf(m, __shfl_xor(m, 2));
        m = fmaxf(m, __shfl_xor(m, 4));
        float s = 0.0f;
        for (int c = sub; c < S_DIM; c += 8) s += expf(row[c] - m);
        s += __shfl_xor(s, 1);
        s += __shfl_xor(s, 2);
        s += __shfl_xor(s, 4);
        if (sub == 0) { rowmax[r] = m; rowrsum[r] = 1.0f / s; }
    }
    __syncthreads();

    // ------- Phase 2b: attn = softmax * groupW -> global (once) + LDS -------
    {
        for (int r = 0; r < ROWS_PER_WG; ++r) {
            const int   gr  = r0 + r;
            const int   grc = gr / Kcls;
            const float m   = rowmax[r];
            const float rs  = rowrsum[r];
            float* srow = sc + r * LDS_STRIDE;
            float* arow = attnp + ((size_t)bidx * S_DIM + gr) * S_DIM;
            for (int c = tid; c < S_DIM; c += 256) {   // coalesced
                float a = expf(srow[c] - m) * rs;
                float w = (gr == c) ? w0 : ((grc == c / Kcls) ? w1 : w2);
                a *= w;
                arow[c] = a;
                srow[c] = a;
            }
        }
    }
    __syncthreads();

    // ---------------- Phase 3: output = attn @ V ----------------
    {
        const int rt   = wave >> 2;          // 2 row tiles
        const int ct   = wave & 3;           // 4 col tiles (D=64)
        const int dcol = ct * 16 + l16;
        const float* vb   = v + (size_t)bidx * S_DIM * D_DIM + dcol;
        const float* arow = sc + (rt * 16 + l16) * LDS_STRIDE + 2 * half;

        v8f c = {};
        for (int kk = 0; kk < S_DIM / 4; ++kk) {
            v2f a = *(const v2f*)(arow + 4 * kk);          // LDS, 8B aligned
            int kr = 4 * kk + 2 * half;
            v2f b;
            b.x = vb[(size_t)kr * D_DIM];                  // V in L2
            b.y = vb[(size_t)(kr + 1) * D_DIM];
            c = __builtin_amdgcn_wmma_f32_16x16x4_f32(
                    false, a, false, b, (short)0, c, false, false);
        }
        #pragma unroll
        for (int i = 0; i < 8; ++i) {
            int m = r0 + rt * 16 + i + 8 * half;
            outp[((size_t)bidx * S_DIM + m) * D_DIM + dcol] = c[i];
        }
    }
}

extern "C" void kernel_launch(void* const* d_in, const int* in_sizes, int n_in,
                              void* d_out, int out_size, void* d_ws, size_t ws_size,
                              hipStream_t stream) {
    const float* q  = (const float*)d_in[0];
    const float* k  = (const float*)d_in[1];
    const float* v  = (const float*)d_in[2];
    const float* aw = (const float*)d_in[3];
    const int*   Kp = (const int*)d_in[5];   // class size K (d_in[4] is N, unused)

    float* outp  = (float*)d_out;
    float* attnp = outp + (size_t)B_DIM * S_DIM * D_DIM;

    const size_t shmem = ((size_t)ROWS_PER_WG * LDS_STRIDE + 64) * sizeof(float);
    // 262,656 B < 320 KB WGP LDS; opt in above the 64KB default.
    hipFuncSetAttribute((const void*)attn_fused_kernel,
                        hipFuncAttributeMaxDynamicSharedMemorySize, (int)shmem);

    dim3 grid(S_DIM / ROWS_PER_WG, B_DIM);   // 64 row-blocks x 32 batches
    attn_fused_kernel<<<grid, 256, shmem, stream>>>(q, k, v, aw, Kp, outp, attnp);
}